// Net_7009386627771
// MI455X (gfx1250) — compile-verified
//
#include <hip/hip_runtime.h>
#include <hip/hip_bf16.h>
#include <math.h>

// ---------------------------------------------------------------------------
// Problem constants (from reference)
// ---------------------------------------------------------------------------
#define BT      2048      // B*T = 8*256
#define IDIM    128
#define ODIM    128
#define HDIM    512
#define CDIM    128
#define PAD     127       // ODIM-1
#define NSHIFT  255       // IDIM + PAD - ... = d + pad = 255 shifts
#define TEMPER  5.0f
#define ENERGY_TH 100.0f
#define N_ITER  4

typedef float v2f __attribute__((ext_vector_type(2)));
typedef float v8f __attribute__((ext_vector_type(8)));

// ---------------------------------------------------------------------------
// init: x_res = x, y_res = y
// ---------------------------------------------------------------------------
__global__ void init_kernel(const float* __restrict__ x, const float* __restrict__ y,
                            float* __restrict__ xres, float* __restrict__ yres, int n) {
    int i = blockIdx.x * blockDim.x + threadIdx.x;
    if (i < n) { xres[i] = x[i]; yres[i] = y[i]; }
}

// ---------------------------------------------------------------------------
// counts: seq-mask row flags + the two global denominators (computed once,
// deterministic single-block fixed-order reduction).
//   scalars[0] = max(#(y!=0), 1)                (energy_loss denom)
//   scalars[1] = max(512 * #(rows not all 0),1) (hsr denomh)
// ---------------------------------------------------------------------------
__global__ __launch_bounds__(256) void counts_kernel(const float* __restrict__ y,
                                                     int* __restrict__ sm_row,
                                                     float* __restrict__ scalars) {
    __shared__ int red[256];
    __shared__ int s_ce;
    int tid = threadIdx.x;
    int ce = 0;
    for (int i = tid; i < BT * ODIM; i += 256) ce += (y[i] != 0.0f) ? 1 : 0;
    int cr = 0;
    for (int r = tid; r < BT; r += 256) {
        const float* yr = y + r * ODIM;
        int allz = 1;
        for (int w = 0; w < ODIM; ++w) if (yr[w] != 0.0f) { allz = 0; break; }
        sm_row[r] = allz;
        cr += allz ? 0 : 1;
    }
    red[tid] = ce; __syncthreads();
    for (int off = 128; off > 0; off >>= 1) { if (tid < off) red[tid] += red[tid + off]; __syncthreads(); }
    if (tid == 0) s_ce = red[0];
    __syncthreads();
    red[tid] = cr; __syncthreads();
    for (int off = 128; off > 0; off >>= 1) { if (tid < off) red[tid] += red[tid + off]; __syncthreads(); }
    if (tid == 0) {
        scalars[0] = fmaxf((float)s_ce, 1.0f);
        scalars[1] = fmaxf((float)(red[0] * HDIM), 1.0f);
    }
}

// ---------------------------------------------------------------------------
// selector: per-row shifted cosine argmax, softmax attention, y_att output,
// and the x_res -= x_ele residual update (x_ele = reverse shift of y_att).
// One block (256 threads) per row.
// ---------------------------------------------------------------------------
__global__ __launch_bounds__(256) void selector_kernel(const float* __restrict__ yres,
                                                       float* __restrict__ xres,
                                                       float* __restrict__ yatt,
                                                       int* __restrict__ theta_out) {
    int row = blockIdx.x;
    int tid = threadIdx.x;
    __shared__ float xp[IDIM + 2 * PAD];   // 382: zero-padded x_res row
    __shared__ float yr[ODIM];
    __shared__ float red[256];
    __shared__ int   redi[256];
    __shared__ float ya[ODIM];
    __shared__ float s_ny, s_max, s_sum;
    __shared__ int   s_theta;

    const float* yrow = yres + row * ODIM;
    float* xrow = xres + row * IDIM;

    if (tid < ODIM) yr[tid] = yrow[tid];
    for (int i = tid; i < IDIM + 2 * PAD; i += 256)
        xp[i] = (i >= PAD && i < PAD + IDIM) ? xrow[i - PAD] : 0.0f;
    __syncthreads();

    // ||y||
    red[tid] = (tid < ODIM) ? yr[tid] * yr[tid] : 0.0f;
    __syncthreads();
    for (int off = 128; off > 0; off >>= 1) { if (tid < off) red[tid] += red[tid + off]; __syncthreads(); }
    if (tid == 0) s_ny = sqrtf(red[0]);
    __syncthreads();
    float ny = s_ny;

    // cosine similarity for shift s = tid (255 shifts)
    float sim = -__builtin_inff();
    if (tid < NSHIFT) {
        float d = 0.0f, n2 = 0.0f;
        for (int w = 0; w < ODIM; ++w) {
            float xv = xp[tid + w];
            d  += xv * yr[w];
            n2 += xv * xv;
        }
        float den = ny * sqrtf(n2);
        sim = (den == 0.0f) ? 0.0f : d / den;
    }
    red[tid] = sim; redi[tid] = tid;
    __syncthreads();
    // argmax with first-occurrence (smallest index) tie-break == jnp.argmax
    for (int off = 128; off > 0; off >>= 1) {
        if (tid < off) {
            float a = red[tid], b = red[tid + off];
            if (b > a || (b == a && redi[tid + off] < redi[tid])) { red[tid] = b; redi[tid] = redi[tid + off]; }
        }
        __syncthreads();
    }
    if (tid == 0) { s_theta = redi[0]; theta_out[row] = redi[0]; }
    __syncthreads();
    int theta = s_theta;

    // softmax(x_sel * y / TEMPER) over 128 dims, y_att = x_sel * attn
    float xsel = 0.0f, z = -__builtin_inff();
    if (tid < ODIM) {
        xsel = xp[theta + tid];
        z = xsel * yr[tid] * (1.0f / TEMPER);
    }
    red[tid] = z; __syncthreads();
    for (int off = 128; off > 0; off >>= 1) { if (tid < off) red[tid] = fmaxf(red[tid], red[tid + off]); __syncthreads(); }
    if (tid == 0) s_max = red[0];
    __syncthreads();
    float ez = (tid < ODIM) ? expf(z - s_max) : 0.0f;
    red[tid] = ez; __syncthreads();
    for (int off = 128; off > 0; off >>= 1) { if (tid < off) red[tid] += red[tid + off]; __syncthreads(); }
    if (tid == 0) s_sum = red[0];
    __syncthreads();
    if (tid < ODIM) {
        float yav = xsel * (ez / s_sum);
        ya[tid] = yav;
        yatt[row * ODIM + tid] = yav;
    }
    __syncthreads();
    // x_ele[w] = y_att[w + 127 - theta] (if in range), x_res -= x_ele
    if (tid < IDIM) {
        int j = tid + PAD - theta;
        float xel = (j >= 0 && j < ODIM) ? ya[j] : 0.0f;
        xrow[tid] -= xel;
    }
}

// ---------------------------------------------------------------------------
// WMMA fp32 GEMM: C(MxN) = A(MxK) @ B(KxN) + bias, using V_WMMA_F32_16X16X4_F32.
// One wave per 16x16 output tile; K accumulated in steps of 4.
// A layout (ISA 32-bit 16x4): lanes 0-15 hold (K=k0,k0+1), lanes 16-31 (k0+2,k0+3).
// B layout is the transpose-symmetric dual; C/D is the 8-VGPR 16x16 layout.
// M, N must be multiples of 16 and the grid must cover tiles exactly.
// ---------------------------------------------------------------------------
__global__ __launch_bounds__(256) void gemm_wmma_f32(const float* __restrict__ A, int lda,
                                                     const float* __restrict__ Bm, int ldb,
                                                     const float* __restrict__ bias,
                                                     float* __restrict__ C, int ldc,
                                                     int M, int N, int K) {
    int lane = threadIdx.x & 31;
    int wave = threadIdx.x >> 5;
    int wgid = blockIdx.x * (blockDim.x >> 5) + wave;
    int ntiles = N >> 4;
    int tm = wgid / ntiles;
    int tn = wgid - tm * ntiles;
    if ((tm << 4) >= M) return;
    int m0 = tm << 4, n0 = tn << 4;
    int half = lane >> 4;        // 0: K pair (0,1), 1: K pair (2,3)
    int l16  = lane & 15;

    v8f acc = {};
    const float* arow = A + (size_t)(m0 + l16) * lda;
    const float* bcol = Bm + n0 + l16;
    for (int k0 = 0; k0 < K; k0 += 4) {
        int ka = k0 + half * 2;
        v2f a, b;
        a[0] = arow[ka];
        a[1] = arow[ka + 1];
        b[0] = bcol[(size_t)ka * ldb];
        b[1] = bcol[(size_t)(ka + 1) * ldb];
        acc = __builtin_amdgcn_wmma_f32_16x16x4_f32(
            /*neg_a=*/false, a, /*neg_b=*/false, b,
            /*c_mod=*/(short)0, acc, /*reuse_a=*/false, /*reuse_b=*/false);
    }
    int nc = n0 + l16;
    float bv = bias[nc];
    int mb = m0 + half * 8;
#pragma unroll
    for (int r = 0; r < 8; ++r)
        C[(size_t)(mb + r) * ldc + nc] = acc[r] + bv;
}

// ---------------------------------------------------------------------------
// hsr: stable top-128-of-512 masking (rank = #(e_j > e_i) + #(j<i, e_j==e_i),
// exactly lax.top_k's index tie-break), mask bookkeeping, loss_h per row.
// One block (256 threads) per row.
// ---------------------------------------------------------------------------
__global__ __launch_bounds__(256) void hsr_kernel(float* __restrict__ h,
                                                  float* __restrict__ mask_prev,
                                                  const int* __restrict__ sm_row,
                                                  float* __restrict__ lossh_rows,
                                                  const float* __restrict__ scalars,
                                                  int iter) {
    int row = blockIdx.x, tid = threadIdx.x;
    __shared__ float e[HDIM];
    __shared__ float hl[HDIM];
    __shared__ float mp[HDIM];
    __shared__ float red[256];
    float* hrow  = h + (size_t)row * HDIM;
    float* mprow = mask_prev + (size_t)row * HDIM;

    for (int i = tid; i < HDIM; i += 256) {
        float hv = hrow[i];
        hl[i] = hv;
        e[i]  = hv * hv;
        mp[i] = (iter > 0) ? mprow[i] : 0.0f;
    }
    __syncthreads();

    int idxs[2] = { tid, tid + 256 };
    if (iter == 0) {
#pragma unroll
        for (int t = 0; t < 2; ++t) {
            int i = idxs[t];
            float ei = e[i];
            int rank = 0;
            for (int j = 0; j < HDIM; ++j) {
                float ej = e[j];
                rank += (ej > ei || (ej == ei && j < i)) ? 1 : 0;
            }
            int mc = rank < CDIM;
            mprow[i] = mc ? 1.0f : 0.0f;
            hrow[i]  = mc ? hl[i] : 0.0f;
        }
        if (tid == 0) lossh_rows[row] = 0.0f;
    } else {
        float denomh = scalars[1];
        int smr = sm_row[row];
        float lsum = 0.0f;
        float hz[2];
#pragma unroll
        for (int t = 0; t < 2; ++t) {
            int i = idxs[t];
            float ei = e[i];
            int rank = 0;
            for (int j = 0; j < HDIM; ++j) {
                float ej = e[j];
                rank += (ej > ei || (ej == ei && j < i)) ? 1 : 0;
            }
            float mc0 = (rank < CDIM) ? 1.0f : 0.0f;
            float minter = mp[i] * mc0;
            if (minter > 0.0f && !smr) {
                float d = hl[i] - (1.0f - minter);
                lsum += d * d / denomh;
            }
            hz[t] = (mp[i] > 0.0f) ? 0.0f : hl[i];
        }
        __syncthreads();
#pragma unroll
        for (int t = 0; t < 2; ++t) e[idxs[t]] = hz[t] * hz[t];
        __syncthreads();
#pragma unroll
        for (int t = 0; t < 2; ++t) {
            int i = idxs[t];
            float ei = e[i];
            int rank = 0;
            for (int j = 0; j < HDIM; ++j) {
                float ej = e[j];
                rank += (ej > ei || (ej == ei && j < i)) ? 1 : 0;
            }
            int mc = rank < CDIM;
            mprow[i] = mp[i] + (mc ? 1.0f : 0.0f);
            hrow[i]  = mc ? hz[t] : 0.0f;
        }
        red[tid] = lsum; __syncthreads();
        for (int off = 128; off > 0; off >>= 1) { if (tid < off) red[tid] += red[tid + off]; __syncthreads(); }
        if (tid == 0) lossh_rows[row] = red[0];
    }
}

// ---------------------------------------------------------------------------
// finish: energy loss per row + y_res -= y_ele. One block (128 threads)/row.
// ---------------------------------------------------------------------------
__global__ __launch_bounds__(128) void finish_kernel(const float* __restrict__ yele,
                                                     float* __restrict__ yres,
                                                     const float* __restrict__ y_in,
                                                     const int* __restrict__ theta,
                                                     const float* __restrict__ scalars,
                                                     float* __restrict__ ll_rows) {
    int row = blockIdx.x, tid = threadIdx.x;
    __shared__ float red[128];
    float denom = scalars[0];
    float th = (float)theta[row];
    float tshift = fabsf(th - (float)(ODIM - 1));
    float move_energy = tshift + 1.0f;
    bool  move_mask = tshift > ENERGY_TH;
    int idx = row * ODIM + tid;
    float ye = yele[idx];
    float yv = yres[idx];
    bool sm = (y_in[idx] == 0.0f);
    float ll = sm ? 0.0f : (ye - yv) * (ye - yv) / denom;
    ll = move_mask ? 0.0f : ll / move_energy;
    red[tid] = ll; __syncthreads();
    for (int off = 64; off > 0; off >>= 1) { if (tid < off) red[tid] += red[tid + off]; __syncthreads(); }
    if (tid == 0) ll_rows[row] = red[0];
    yres[idx] = yv - ye;
}

// ---------------------------------------------------------------------------
// per-iteration deterministic reduction of row losses -> loss_acc[iter]
// ---------------------------------------------------------------------------
__global__ __launch_bounds__(256) void reduce_kernel(const float* __restrict__ lossh_rows,
                                                     const float* __restrict__ ll_rows,
                                                     float* __restrict__ loss_acc, int iter) {
    __shared__ float red[256];
    int tid = threadIdx.x;
    float s = 0.0f;
    for (int r = tid; r < BT; r += 256) s += lossh_rows[r] + ll_rows[r];
    red[tid] = s; __syncthreads();
    for (int off = 128; off > 0; off >>= 1) { if (tid < off) red[tid] += red[tid + off]; __syncthreads(); }
    if (tid == 0) loss_acc[iter] = red[0];
}

__global__ void out_kernel(const float* __restrict__ loss_acc, float* __restrict__ out) {
    if (threadIdx.x == 0 && blockIdx.x == 0)
        out[0] = (loss_acc[0] + loss_acc[1] + loss_acc[2] + loss_acc[3]) * 0.25f;
}

// ---------------------------------------------------------------------------
// launch
// ---------------------------------------------------------------------------
extern "C" void kernel_launch(void* const* d_in, const int* in_sizes, int n_in,
                              void* d_out, int out_size, void* d_ws, size_t ws_size,
                              hipStream_t stream) {
    const float* x     = (const float*)d_in[0];
    const float* y     = (const float*)d_in[1];
    const float* enc_w = (const float*)d_in[2];   // 128 x 512
    const float* enc_b = (const float*)d_in[3];   // 512
    const float* dec_w = (const float*)d_in[4];   // 512 x 128
    const float* dec_b = (const float*)d_in[5];   // 128
    float* out = (float*)d_out;

    float* ws = (float*)d_ws;
    size_t off = 0;
    float* xres      = ws + off; off += (size_t)BT * IDIM;   // 262144
    float* yres      = ws + off; off += (size_t)BT * ODIM;
    float* yatt      = ws + off; off += (size_t)BT * ODIM;
    float* yele      = ws + off; off += (size_t)BT * ODIM;
    float* h         = ws + off; off += (size_t)BT * HDIM;   // 1048576
    float* mask_prev = ws + off; off += (size_t)BT * HDIM;
    float* lossh_rows= ws + off; off += BT;
    float* ll_rows   = ws + off; off += BT;
    float* loss_acc  = ws + off; off += 4;
    float* scalars   = ws + off; off += 4;                   // [denom, denomh]
    int*   theta     = (int*)(ws + off); off += BT;
    int*   sm_row    = (int*)(ws + off); off += BT;

    init_kernel<<<(BT * IDIM + 255) / 256, 256, 0, stream>>>(x, y, xres, yres, BT * IDIM);
    counts_kernel<<<1, 256, 0, stream>>>(y, sm_row, scalars);

    for (int iter = 0; iter < N_ITER; ++iter) {
        selector_kernel<<<BT, 256, 0, stream>>>(yres, xres, yatt, theta);
        // h = y_att(2048x128) @ enc_w(128x512) + enc_b : 128x32 tiles, 8 waves/block
        gemm_wmma_f32<<<(128 * 32) / 8, 256, 0, stream>>>(yatt, IDIM, enc_w, HDIM, enc_b,
                                                          h, HDIM, BT, HDIM, IDIM);
        hsr_kernel<<<BT, 256, 0, stream>>>(h, mask_prev, sm_row, lossh_rows, scalars, iter);
        // y_ele = h(2048x512) @ dec_w(512x128) + dec_b : 128x8 tiles, 8 waves/block
        gemm_wmma_f32<<<(128 * 8) / 8, 256, 0, stream>>>(h, HDIM, dec_w, ODIM, dec_b,
                                                         yele, ODIM, BT, ODIM, HDIM);
        finish_kernel<<<BT, 128, 0, stream>>>(yele, yres, y, theta, scalars, ll_rows);
        reduce_kernel<<<1, 256, 0, stream>>>(lossh_rows, ll_rows, loss_acc, iter);
    }
    out_kernel<<<1, 32, 0, stream>>>(loss_acc, out);
}